// nmODEBlock_51591147159877
// MI455X (gfx1250) — compile-verified
//
#include <hip/hip_runtime.h>

// nmODE: out = relu(p(1)) where dp/dt = -p + sin^2(p + gamma), p(0)=0, gamma = x.
// gamma -> p(1) is pi-periodic (sin^2 is pi-periodic) and smooth, so tabulate it
// once (fp64 dopri5, 8192 samples = 32 KB) and the 64M-element main pass is a
// memory-bound LDS-LUT lerp: ~22us HBM floor (512 MB @ 23.3 TB/s) instead of
// ~1ms of per-element transcendentals (192 sins/element).

#define LUT_BITS 13
#define LUT_N    (1 << LUT_BITS)      // 8192 entries = 32 KB
#define LUT_MASK (LUT_N - 1)
#define N_STEPS  32

typedef float __attribute__((ext_vector_type(4))) floatx4;

static __device__ __forceinline__ double f_rhs(double p, double gamma) {
    double s = sin(p + gamma);
    return s * s - p;
}

// ---------------------------------------------------------------------------
// Kernel 1: build the pi-periodic LUT in fp64 (8192 x 32 dopri5 steps; trivial)
// ---------------------------------------------------------------------------
__global__ void nmode_build_lut(float* __restrict__ lut) {
    int i = blockIdx.x * blockDim.x + threadIdx.x;
    if (i >= LUT_N) return;
    const double PI_D = 3.14159265358979323846;
    double gamma = (double)i * (PI_D / (double)LUT_N);
    double p = 0.0;
    const double h = 1.0 / (double)N_STEPS;
    for (int s = 0; s < N_STEPS; ++s) {
        double k1 = f_rhs(p, gamma);
        double k2 = f_rhs(p + h * (1.0/5.0) * k1, gamma);
        double k3 = f_rhs(p + h * (3.0/40.0*k1 + 9.0/40.0*k2), gamma);
        double k4 = f_rhs(p + h * (44.0/45.0*k1 - 56.0/15.0*k2 + 32.0/9.0*k3), gamma);
        double k5 = f_rhs(p + h * (19372.0/6561.0*k1 - 25360.0/2187.0*k2
                                   + 64448.0/6561.0*k3 - 212.0/729.0*k4), gamma);
        double k6 = f_rhs(p + h * (9017.0/3168.0*k1 - 355.0/33.0*k2
                                   + 46732.0/5247.0*k3 + 49.0/176.0*k4
                                   - 5103.0/18656.0*k5), gamma);
        p = p + h * (35.0/384.0*k1 + 500.0/1113.0*k3 + 125.0/192.0*k4
                     - 2187.0/6784.0*k5 + 11.0/84.0*k6);
    }
    lut[i] = (float)p;
}

// ---------------------------------------------------------------------------
// LUT lookup: frac(gamma/pi) -> lerp between adjacent LDS entries -> relu.
// ---------------------------------------------------------------------------
static __device__ __forceinline__ float lut_eval(const float* lut, float g) {
    const float INV_PI = 0.31830988618379067154f;
    float u = g * INV_PI;
    u -= floorf(u);                        // frac in [0,1)
    float fid = u * (float)LUT_N;
    int i0 = (int)fid;
    float w = fid - (float)i0;
    i0 &= LUT_MASK;
    int i1 = (i0 + 1) & LUT_MASK;
    float a = lut[i0];
    float b = lut[i1];
    return fmaxf(fmaf(w, b - a, a), 0.0f);
}

// ---------------------------------------------------------------------------
// Kernel 2: streaming pass. Stage LUT -> LDS with gfx1250 async-to-LDS DMA,
// then NT b128 load -> LDS lerp -> relu -> NT b128 store (2-wide MLP unroll).
// ---------------------------------------------------------------------------
__global__ void __launch_bounds__(256) nmode_apply(const float* __restrict__ x,
                                                   const float* __restrict__ lutG,
                                                   float* __restrict__ out,
                                                   int n4, int tail) {
    __shared__ float lut[LUT_N];

    // Async copy 32 KB LUT global->LDS (ASYNCcnt path, no VGPR staging).
    {
        unsigned ldsBase = (unsigned)(uintptr_t)(void*)lut;   // low 32 bits = LDS offset
        for (int i = threadIdx.x; i < LUT_N / 4; i += blockDim.x) {
            unsigned ldsAddr = ldsBase + (unsigned)i * 16u;
            unsigned long long gAddr =
                (unsigned long long)(uintptr_t)(lutG + 4 * i);
            asm volatile("global_load_async_to_lds_b128 %0, %1, off"
                         :: "v"(ldsAddr), "v"(gAddr) : "memory");
        }
        asm volatile("s_wait_asynccnt 0" ::: "memory");
    }
    __syncthreads();

    const floatx4* x4 = (const floatx4*)x;
    floatx4*       o4 = (floatx4*)out;
    int stride = gridDim.x * blockDim.x;
    int tid0   = blockIdx.x * blockDim.x + threadIdx.x;

    // 2-wide unrolled grid-stride loop: two independent NT b128 loads in
    // flight before the dependent VALU/DS work.
    for (int idx = tid0; idx < n4; idx += 2 * stride) {
        int idxB  = idx + stride;
        bool hasB = idxB < n4;
        floatx4 v0 = __builtin_nontemporal_load(&x4[idx]);
        floatx4 v1;
        if (hasB) v1 = __builtin_nontemporal_load(&x4[idxB]);

        floatx4 q0;
        q0.x = lut_eval(lut, v0.x);
        q0.y = lut_eval(lut, v0.y);
        q0.z = lut_eval(lut, v0.z);
        q0.w = lut_eval(lut, v0.w);
        __builtin_nontemporal_store(q0, &o4[idx]);

        if (hasB) {
            floatx4 q1;
            q1.x = lut_eval(lut, v1.x);
            q1.y = lut_eval(lut, v1.y);
            q1.z = lut_eval(lut, v1.z);
            q1.w = lut_eval(lut, v1.w);
            __builtin_nontemporal_store(q1, &o4[idxB]);
        }
    }

    // Scalar tail (n not multiple of 4) — not hit for the given shape.
    if (tid0 < tail) {
        int idx = n4 * 4 + tid0;
        out[idx] = lut_eval(lut, x[idx]);
    }
}

// ---------------------------------------------------------------------------
// Fallback: direct fixed dopri5 per element using hardware v_cos_f32 via
// sin^2(u) = 0.5 - 0.5*cos(2u);  cos(2u) = V_COS((p+gamma)/pi)  (rev units).
// Only used if the workspace cannot hold the LUT.
// ---------------------------------------------------------------------------
static __device__ __forceinline__ float feval_fast(float p, float g_over_pi) {
    float c = __builtin_amdgcn_cosf(fmaf(p, 0.31830988618379067154f, g_over_pi));
    return fmaf(-0.5f, c, 0.5f) - p;
}

__global__ void __launch_bounds__(256) nmode_direct(const float* __restrict__ x,
                                                    float* __restrict__ out, int n) {
    int stride = gridDim.x * blockDim.x;
    for (int idx = blockIdx.x * blockDim.x + threadIdx.x; idx < n; idx += stride) {
        float g = x[idx] * 0.31830988618379067154f;   // gamma/pi
        float p = 0.0f;
        const float h = 1.0f / (float)N_STEPS;
        for (int s = 0; s < N_STEPS; ++s) {
            float k1 = feval_fast(p, g);
            float k2 = feval_fast(fmaf(h * (1.0f/5.0f), k1, p), g);
            float k3 = feval_fast(p + h * (3.0f/40.0f*k1 + 9.0f/40.0f*k2), g);
            float k4 = feval_fast(p + h * (44.0f/45.0f*k1 - 56.0f/15.0f*k2
                                           + 32.0f/9.0f*k3), g);
            float k5 = feval_fast(p + h * (19372.0f/6561.0f*k1 - 25360.0f/2187.0f*k2
                                           + 64448.0f/6561.0f*k3 - 212.0f/729.0f*k4), g);
            float k6 = feval_fast(p + h * (9017.0f/3168.0f*k1 - 355.0f/33.0f*k2
                                           + 46732.0f/5247.0f*k3 + 49.0f/176.0f*k4
                                           - 5103.0f/18656.0f*k5), g);
            p = p + h * (35.0f/384.0f*k1 + 500.0f/1113.0f*k3 + 125.0f/192.0f*k4
                         - 2187.0f/6784.0f*k5 + 11.0f/84.0f*k6);
        }
        out[idx] = fmaxf(p, 0.0f);
    }
}

// ---------------------------------------------------------------------------
extern "C" void kernel_launch(void* const* d_in, const int* in_sizes, int n_in,
                              void* d_out, int out_size, void* d_ws, size_t ws_size,
                              hipStream_t stream) {
    const float* x   = (const float*)d_in[0];
    float*       out = (float*)d_out;
    int n = out_size;

    if (ws_size >= (size_t)LUT_N * sizeof(float)) {
        float* lutG = (float*)d_ws;
        nmode_build_lut<<<LUT_N / 256, 256, 0, stream>>>(lutG);
        int n4   = n >> 2;
        int tail = n & 3;
        int blocks = 2048;                 // ~32 float4 per thread on the full tensor
        nmode_apply<<<blocks, 256, 0, stream>>>(x, lutG, out, n4, tail);
    } else {
        nmode_direct<<<4096, 256, 0, stream>>>(x, out, n);
    }
}